// TabR_82154134437918
// MI455X (gfx1250) — compile-verified
//
#include <hip/hip_runtime.h>
#include <math.h>

// TabR pipeline for MI455X (gfx1250, wave32, WMMA).
// All GEMMs use V_WMMA_F32_16X16X4_F32 (exact fp32 matrix path).
//
// Stages:
//  k_pack   : one-time repack of row-major weights into WMMA-B-fragment
//             layout so each B fragment is a single global_load_b64
//  k_encode : X@lin_W+lin_b -> h ; h@K_W+K_b -> k ; row sumsq   (WMMA x2)
//  k_knn    : per (16-row group, N-segment) score tiles via WMMA into LDS,
//             exact top-96 per segment via wave-cooperative replace-min,
//             global_prefetch of next candidate chunk
//  k_merge  : exact merge of 8 segment top-96 lists -> global top-96
//  k_final  : softmax(scores) ; g = sum_c p_c*relu((k-ck_c)@T_W1+b1) via WMMA
//             (T_W2 folded AFTER the weighted sum: 2x FLOP savings),
//             residual + prenorm MLP block + head, all per-row in one block.

#define D_IN    64
#define D_MAIN  256
#define D_BLOCK 512
#define BQ      1024
#define NC      100000
#define CTX     96
#define PSEG    8
#define NSEG    (NC / PSEG)      // 12500
#define NEG_INF (-3.0e38f)

typedef __attribute__((ext_vector_type(2))) float v2f;
typedef __attribute__((ext_vector_type(8))) float v8f;

__device__ __forceinline__ v8f wmma_f32(v2f a, v2f b, v8f c) {
  // D = A(16x4) * B(4x16) + C(16x16), fp32
  return __builtin_amdgcn_wmma_f32_16x16x4_f32(false, a, false, b, (short)0, c,
                                               false, false);
}

// A-fragment from LDS: lane holds row m, K columns kc..kc+1 (kc = kk*4 + 2*hi)
__device__ __forceinline__ v2f ldsA(const float* s, int stride, int m, int kc) {
  v2f a; a.x = s[m * stride + kc]; a.y = s[m * stride + kc + 1]; return a;
}
// B-fragment from packed weights: single aligned b64 load.
// Packed layout: Wp[((kk*2+hi)*ldw + col)*2 + j] = W[kk*4+2*hi+j][col]
__device__ __forceinline__ v2f packB(const float* __restrict__ Wp, int ldw,
                                     int kk, int hi, int col) {
  return *(const v2f*)(Wp + (((size_t)(kk * 2 + hi) * ldw + col) << 1));
}

// ---------------------------------------------------------------------------
// One-time weight repack into B-fragment layout.
// ---------------------------------------------------------------------------
__global__ void k_pack(const float* __restrict__ W, float* __restrict__ Wp,
                       int K, int ldw) {
  int idx = blockIdx.x * 256 + threadIdx.x;
  if (idx >= K * ldw) return;
  int kr = idx / ldw, col = idx - kr * ldw;
  int kk = kr >> 2, hi = (kr >> 1) & 1, j = kr & 1;
  Wp[((((size_t)(kk * 2 + hi)) * ldw + col) << 1) + j] = W[idx];
}

// ---------------------------------------------------------------------------
// Encoder: out_k = (X@linW+linb)@KW+Kb ; out_sq = rowwise ||out_k||^2 ;
// optionally out_h = X@linW+linb. 16 rows per workgroup, 256 threads (8 waves).
// Weight operands come pre-packed (linWp, KWp).
// ---------------------------------------------------------------------------
__global__ void k_encode(const float* __restrict__ X,
                         const float* __restrict__ linWp, const float* __restrict__ linb,
                         const float* __restrict__ KWp,  const float* __restrict__ Kb,
                         float* __restrict__ outK, float* __restrict__ outSq,
                         float* __restrict__ outH) {
  __shared__ float xs[16 * 68];    // 16 x 64, padded stride 68
  __shared__ float hs[16 * 260];   // 16 x 256, padded stride 260
  __shared__ float cks[16 * 260];
  const int tid  = threadIdx.x;
  const int wave = tid >> 5, lane = tid & 31;
  const int cl   = lane & 15, hi = lane >> 4;
  const int base = blockIdx.x * 16;

  for (int e = tid; e < 16 * D_IN; e += 256) {
    int r = e >> 6, d = e & 63;
    xs[r * 68 + d] = X[(size_t)(base + r) * D_IN + d];
  }
  __syncthreads();

  // h = X @ linW + linb   (K=64 -> 16 k-steps; 16 n-tiles over 8 waves)
  for (int t = 0; t < 2; ++t) {
    int nt = wave * 2 + t;
    int col = nt * 16 + cl;
    v8f acc = {0.f,0.f,0.f,0.f,0.f,0.f,0.f,0.f};
    for (int kk = 0; kk < D_IN / 4; ++kk) {
      v2f a = ldsA(xs, 68, cl, kk * 4 + 2 * hi);
      v2f b = packB(linWp, D_MAIN, kk, hi, col);
      acc = wmma_f32(a, b, acc);
    }
    float bb = linb[col];
#pragma unroll
    for (int r = 0; r < 8; ++r)
      hs[(r + 8 * hi) * 260 + col] = acc[r] + bb;
  }
  __syncthreads();

  // ck = h @ KW + Kb   (K=256 -> 64 k-steps)
  for (int t = 0; t < 2; ++t) {
    int nt = wave * 2 + t;
    int col = nt * 16 + cl;
    v8f acc = {0.f,0.f,0.f,0.f,0.f,0.f,0.f,0.f};
    for (int kk = 0; kk < D_MAIN / 4; ++kk) {
      v2f a = ldsA(hs, 260, cl, kk * 4 + 2 * hi);
      v2f b = packB(KWp, D_MAIN, kk, hi, col);
      acc = wmma_f32(a, b, acc);
    }
    float bb = Kb[col];
#pragma unroll
    for (int r = 0; r < 8; ++r)
      cks[(r + 8 * hi) * 260 + col] = acc[r] + bb;
  }
  __syncthreads();

  for (int it = 0; it < 16; ++it)
    outK[(size_t)(base + it) * D_MAIN + tid] = cks[it * 260 + tid];
  if (outH)
    for (int it = 0; it < 16; ++it)
      outH[(size_t)(base + it) * D_MAIN + tid] = hs[it * 260 + tid];
  if (tid < 16) {
    float s = 0.f;
    for (int d = 0; d < D_MAIN; ++d) { float v = cks[tid * 260 + d]; s += v * v; }
    outSq[base + tid] = s;
  }
}

// ---------------------------------------------------------------------------
// Wave-cooperative exact top-K maintenance (replace-min). One wave owns one
// list. 96 = 3*32 entries; min/argmin recomputed wave-parallel on insert.
// ---------------------------------------------------------------------------
__device__ __forceinline__ void topk_insert(float s, int gi,
                                            volatile float* tv, volatile int* ti,
                                            volatile float* cmin, volatile int* cpos,
                                            int lane) {
  unsigned long long m = __ballot(s > *cmin);
  while (m) {
    int src = __ffsll(m) - 1;
    m &= m - 1;
    float sv = __shfl(s, src);
    int   iv = __shfl(gi, src);
    if (sv > *cmin) {
      int pos = *cpos;
      tv[pos] = sv; ti[pos] = iv;
      float v = tv[lane]; int p = lane;
      float v1 = tv[lane + 32]; if (v1 < v) { v = v1; p = lane + 32; }
      float v2 = tv[lane + 64]; if (v2 < v) { v = v2; p = lane + 64; }
#pragma unroll
      for (int off = 16; off > 0; off >>= 1) {
        float ov = __shfl_xor(v, off); int op = __shfl_xor(p, off);
        if (ov < v || (ov == v && op < p)) { v = ov; p = op; }
      }
      if (lane == 0) { *cmin = v; *cpos = p; }
    }
  }
}

// ---------------------------------------------------------------------------
// kNN: grid (BQ/16, PSEG). Each workgroup: 16 query rows x one N-segment.
// Scores = 2*k.ck - |k|^2 - |ck|^2 via WMMA tiles into LDS; exact per-segment
// top-96 per row. Next chunk of cand_k is prefetched during top-k phase.
// ---------------------------------------------------------------------------
__global__ void k_knn(const float* __restrict__ xk, const float* __restrict__ ksq,
                      const float* __restrict__ candk, const float* __restrict__ csq,
                      float* __restrict__ pv, int* __restrict__ pi) {
  __shared__ float ka[16 * 260];
  __shared__ float sc[16 * 260];
  __shared__ float tv[16][CTX];
  __shared__ int   ti[16][CTX];
  __shared__ float cmin[16];
  __shared__ int   cpos[16];
  __shared__ float kq[16];
  const int tid  = threadIdx.x, wave = tid >> 5, lane = tid & 31;
  const int cl   = lane & 15, hi = lane >> 4;
  const int bg   = blockIdx.x;
  const int seg  = blockIdx.y;
  const int base = seg * NSEG;

  for (int it = 0; it < 16; ++it)
    ka[it * 260 + tid] = xk[(size_t)(bg * 16 + it) * D_MAIN + tid];
  if (tid < 16) {
    kq[tid] = ksq[bg * 16 + tid];
    cmin[tid] = NEG_INF; cpos[tid] = 0;
  }
  for (int e = tid; e < 16 * CTX; e += 256) {
    tv[e / CTX][e % CTX] = NEG_INF;
    ti[e / CTX][e % CTX] = 0;
  }
  __syncthreads();

  const int nch = (NSEG + 255) / 256;  // 49
  for (int ch = 0; ch < nch; ++ch) {
    // --- score phase: 16 col-tiles / 8 waves
    for (int t = 0; t < 2; ++t) {
      int nt   = wave * 2 + t;
      int cc   = nt * 16 + cl;        // col within 256-chunk
      int lcol = ch * 256 + cc;       // col within segment
      bool valid = lcol < NSEG;
      int gn = valid ? (base + lcol) : 0;
      const float* bp = candk + (size_t)gn * D_MAIN + 2 * hi;
      v8f acc = {0.f,0.f,0.f,0.f,0.f,0.f,0.f,0.f};
      for (int kk = 0; kk < D_MAIN / 4; ++kk) {
        v2f a = ldsA(ka, 260, cl, kk * 4 + 2 * hi);
        v2f b = *(const v2f*)(bp + kk * 4);   // contiguous pair: one b64
        acc = wmma_f32(a, b, acc);
      }
      float cq = valid ? csq[gn] : 0.f;
#pragma unroll
      for (int r = 0; r < 8; ++r) {
        int row = r + 8 * hi;
        float s = valid ? (2.f * acc[r] - kq[row] - cq) : NEG_INF;
        sc[row * 260 + cc] = s;
      }
    }
    // prefetch next chunk's candidate rows toward this WGP (global_prefetch_b8)
    {
      int lnext = (ch + 1) * 256 + tid;
      if (lnext < NSEG)
        __builtin_prefetch(candk + (size_t)(base + lnext) * D_MAIN, 0, 3);
    }
    __syncthreads();
    // --- top-k phase: wave w owns rows 2w, 2w+1
    for (int rr = 0; rr < 2; ++rr) {
      int row = wave * 2 + rr;
      for (int it = 0; it < 8; ++it) {
        int cidx = it * 32 + lane;
        float s  = sc[row * 260 + cidx];
        int  gi  = base + ch * 256 + cidx;
        topk_insert(s, gi, tv[row], ti[row], &cmin[row], &cpos[row], lane);
      }
    }
    __syncthreads();
  }

  for (int e = tid; e < 16 * CTX; e += 256) {
    int r = e / CTX, j = e % CTX;
    size_t o = ((size_t)(bg * 16 + r) * PSEG + seg) * CTX + j;
    pv[o] = tv[r][j];
    pi[o] = ti[r][j];
  }
}

// ---------------------------------------------------------------------------
// Merge PSEG partial top-96 lists per row into the exact global top-96.
// One wave per row.
// ---------------------------------------------------------------------------
__global__ void k_merge(const float* __restrict__ pv, const int* __restrict__ pi,
                        float* __restrict__ tvout, int* __restrict__ tiout) {
  __shared__ float tv[CTX];
  __shared__ int   ti[CTX];
  __shared__ float cmin[1];
  __shared__ int   cpos[1];
  const int lane = threadIdx.x;
  const int b = blockIdx.x;
  tv[lane] = NEG_INF; tv[lane + 32] = NEG_INF; tv[lane + 64] = NEG_INF;
  ti[lane] = 0; ti[lane + 32] = 0; ti[lane + 64] = 0;
  if (lane == 0) { cmin[0] = NEG_INF; cpos[0] = 0; }
  __syncthreads();
  const int M = PSEG * CTX;  // 768
  for (int it = 0; it < M / 32; ++it) {
    int j = it * 32 + lane;
    float s = pv[(size_t)b * M + j];
    int  gi = pi[(size_t)b * M + j];
    topk_insert(s, gi, tv, ti, cmin, cpos, lane);
  }
  __syncthreads();
  tvout[b * CTX + lane]      = tv[lane];
  tvout[b * CTX + lane + 32] = tv[lane + 32];
  tvout[b * CTX + lane + 64] = tv[lane + 64];
  tiout[b * CTX + lane]      = ti[lane];
  tiout[b * CTX + lane + 32] = ti[lane + 32];
  tiout[b * CTX + lane + 64] = ti[lane + 64];
}

__device__ __forceinline__ float block_sum(float v, volatile float* red, int tid) {
  __syncthreads();
  red[tid] = v;
  __syncthreads();
  for (int s = 128; s > 0; s >>= 1) {
    if (tid < s) red[tid] += red[tid + s];
    __syncthreads();
  }
  float r = red[0];
  __syncthreads();
  return r;
}

// ---------------------------------------------------------------------------
// Final: softmax over stored scores; g = sum_c p_c * relu((k-ck_c)@TW1+Tb1)
// via WMMA (16 ctx rows at a time); ctx = sy*ye_w+ye_b + g@TW2 ; residual;
// prenorm MLP block; head. One query row per workgroup, 256 threads.
// TW1 comes pre-packed (TW1p).
// ---------------------------------------------------------------------------
__global__ void k_final(const float* __restrict__ xh, const float* __restrict__ xk,
                        const float* __restrict__ tv, const int* __restrict__ ti,
                        const float* __restrict__ candy, const float* __restrict__ candk,
                        const float* __restrict__ yew, const float* __restrict__ yeb,
                        const float* __restrict__ TW1p, const float* __restrict__ Tb1,
                        const float* __restrict__ TW2,
                        const float* __restrict__ bg, const float* __restrict__ bbe,
                        const float* __restrict__ bW1, const float* __restrict__ bb1,
                        const float* __restrict__ bW2, const float* __restrict__ bb2,
                        const float* __restrict__ hg, const float* __restrict__ hbe,
                        const float* __restrict__ hW, const float* __restrict__ hb,
                        float* __restrict__ out) {
  __shared__ float kx[D_MAIN];
  __shared__ float xe[D_MAIN];
  __shared__ float scl[CTX];
  __shared__ int   til[CTX];
  __shared__ float probs[CTX];
  __shared__ float difft[16 * 260];
  __shared__ float g[D_BLOCK];
  __shared__ float xbuf[D_MAIN];
  __shared__ float zbuf[D_MAIN];
  __shared__ float h2[D_BLOCK];
  __shared__ float red[256];
  __shared__ float syv;

  const int tid  = threadIdx.x, wave = tid >> 5, lane = tid & 31;
  const int cl   = lane & 15, hi = lane >> 4;
  const int b    = blockIdx.x;

  kx[tid] = xk[(size_t)b * D_MAIN + tid];
  xe[tid] = xh[(size_t)b * D_MAIN + tid];
  if (tid < CTX) { scl[tid] = tv[b * CTX + tid]; til[tid] = ti[b * CTX + tid]; }
  __syncthreads();

  if (wave == 0) {  // softmax + sum_c p_c * y_c   (96 = 3*32)
    float a0 = scl[lane], a1 = scl[lane + 32], a2 = scl[lane + 64];
    float mx = fmaxf(a0, fmaxf(a1, a2));
#pragma unroll
    for (int off = 16; off > 0; off >>= 1) mx = fmaxf(mx, __shfl_xor(mx, off));
    float e0 = __expf(a0 - mx), e1 = __expf(a1 - mx), e2 = __expf(a2 - mx);
    float sm = e0 + e1 + e2;
#pragma unroll
    for (int off = 16; off > 0; off >>= 1) sm += __shfl_xor(sm, off);
    float inv = 1.f / sm;
    probs[lane] = e0 * inv; probs[lane + 32] = e1 * inv; probs[lane + 64] = e2 * inv;
    float sy = e0 * inv * candy[til[lane]] + e1 * inv * candy[til[lane + 32]] +
               e2 * inv * candy[til[lane + 64]];
#pragma unroll
    for (int off = 16; off > 0; off >>= 1) sy += __shfl_xor(sy, off);
    if (lane == 0) syv = sy;
  }
  __syncthreads();

  // g[n] = sum_c p_c * relu(diff_c @ TW1 + Tb1)[n]  (WMMA, ctx tiles of 16)
  float gacc[4] = {0.f, 0.f, 0.f, 0.f};
  for (int ct = 0; ct < CTX / 16; ++ct) {
    for (int it = 0; it < 16; ++it) {
      int c = ct * 16 + it;
      difft[it * 260 + tid] = kx[tid] - candk[(size_t)til[c] * D_MAIN + tid];
    }
    __syncthreads();
#pragma unroll
    for (int q = 0; q < 4; ++q) {
      int nt = wave + q * 8;       // wave owns 4 of 32 n-tiles
      int col = nt * 16 + cl;
      v8f acc = {0.f,0.f,0.f,0.f,0.f,0.f,0.f,0.f};
      for (int kk = 0; kk < D_MAIN / 4; ++kk) {
        v2f a  = ldsA(difft, 260, cl, kk * 4 + 2 * hi);
        v2f bv = packB(TW1p, D_BLOCK, kk, hi, col);
        acc = wmma_f32(a, bv, acc);
      }
      float bb = Tb1[col];
#pragma unroll
      for (int r = 0; r < 8; ++r) {
        float h = fmaxf(acc[r] + bb, 0.f);
        gacc[q] += probs[ct * 16 + r + 8 * hi] * h;
      }
    }
    __syncthreads();
  }
#pragma unroll
  for (int q = 0; q < 4; ++q) {
    float t = gacc[q] + __shfl_xor(gacc[q], 16);
    if (hi == 0) g[(wave + q * 8) * 16 + cl] = t;
  }
  __syncthreads();

  // context_x + residual (TW2 folded after the prob-weighted sum)
  {
    float ctxv = syv * yew[tid] + yeb[tid];
    for (int j = 0; j < D_BLOCK; ++j) ctxv += g[j] * TW2[(size_t)j * D_MAIN + tid];
    xbuf[tid] = xe[tid] + ctxv;
  }
  __syncthreads();

  // prenorm residual block
  float xv   = xbuf[tid];
  float mean = block_sum(xv, red, tid) * (1.f / 256.f);
  float dv   = xv - mean;
  float var  = block_sum(dv * dv, red, tid) * (1.f / 256.f);
  zbuf[tid]  = dv * rsqrtf(var + 1e-5f) * bg[tid] + bbe[tid];
  __syncthreads();
  for (int t = 0; t < 2; ++t) {
    int j = tid + t * 256;
    float a = bb1[j];
    for (int d = 0; d < D_MAIN; ++d) a += zbuf[d] * bW1[(size_t)d * D_BLOCK + j];
    h2[j] = fmaxf(a, 0.f);
  }
  __syncthreads();
  {
    float a = bb2[tid];
    for (int j = 0; j < D_BLOCK; ++j) a += h2[j] * bW2[(size_t)j * D_MAIN + tid];
    xbuf[tid] = xbuf[tid] + a;
  }
  __syncthreads();

  // head: relu(LN(x)) @ head_W + head_b
  float x2 = xbuf[tid];
  float m2 = block_sum(x2, red, tid) * (1.f / 256.f);
  float d2 = x2 - m2;
  float v2 = block_sum(d2 * d2, red, tid) * (1.f / 256.f);
  float z2 = d2 * rsqrtf(v2 + 1e-5f) * hg[tid] + hbe[tid];
  float r2 = fmaxf(z2, 0.f);
  float dot = block_sum(r2 * hW[tid], red, tid);
  if (tid == 0) out[b] = dot + hb[0];
}

// ---------------------------------------------------------------------------
extern "C" void kernel_launch(void* const* d_in, const int* in_sizes, int n_in,
                              void* d_out, int out_size, void* d_ws, size_t ws_size,
                              hipStream_t stream) {
  const float* x_num = (const float*)d_in[0];
  const float* cxn   = (const float*)d_in[1];
  const float* candy = (const float*)d_in[2];
  const float* linW  = (const float*)d_in[3];
  const float* linb  = (const float*)d_in[4];
  const float* KW    = (const float*)d_in[5];
  const float* Kb    = (const float*)d_in[6];
  const float* yew   = (const float*)d_in[7];
  const float* yeb   = (const float*)d_in[8];
  const float* TW1   = (const float*)d_in[9];
  const float* Tb1   = (const float*)d_in[10];
  const float* TW2   = (const float*)d_in[11];
  const float* bgp   = (const float*)d_in[12];
  const float* bbe   = (const float*)d_in[13];
  const float* bW1   = (const float*)d_in[14];
  const float* bb1   = (const float*)d_in[15];
  const float* bW2   = (const float*)d_in[16];
  const float* bb2   = (const float*)d_in[17];
  const float* hg    = (const float*)d_in[18];
  const float* hbe   = (const float*)d_in[19];
  const float* hW    = (const float*)d_in[20];
  const float* hb    = (const float*)d_in[21];
  (void)in_sizes; (void)n_in; (void)out_size; (void)ws_size;

  char* ws = (char*)d_ws;
  size_t off = 0;
  auto carve = [&](size_t bytes) {
    void* p = ws + off;
    off = (off + bytes + 255) & ~(size_t)255;
    return p;
  };
  float* cand_k = (float*)carve((size_t)NC * D_MAIN * sizeof(float));   // 102.4 MB (L2-resident)
  float* c_sq   = (float*)carve((size_t)NC * sizeof(float));
  float* xhbuf  = (float*)carve((size_t)BQ * D_MAIN * sizeof(float));
  float* xkbuf  = (float*)carve((size_t)BQ * D_MAIN * sizeof(float));
  float* k_sq   = (float*)carve((size_t)BQ * sizeof(float));
  float* part_v = (float*)carve((size_t)BQ * PSEG * CTX * sizeof(float));
  int*   part_i = (int*)carve((size_t)BQ * PSEG * CTX * sizeof(int));
  float* top_v  = (float*)carve((size_t)BQ * CTX * sizeof(float));
  int*   top_i  = (int*)carve((size_t)BQ * CTX * sizeof(int));
  float* linWp  = (float*)carve((size_t)D_IN * D_MAIN * sizeof(float));
  float* KWp    = (float*)carve((size_t)D_MAIN * D_MAIN * sizeof(float));
  float* TW1p   = (float*)carve((size_t)D_MAIN * D_BLOCK * sizeof(float));

  // one-time weight repack into WMMA B-fragment layout
  k_pack<<<(D_IN * D_MAIN + 255) / 256, 256, 0, stream>>>(linW, linWp, D_IN, D_MAIN);
  k_pack<<<(D_MAIN * D_MAIN + 255) / 256, 256, 0, stream>>>(KW, KWp, D_MAIN, D_MAIN);
  k_pack<<<(D_MAIN * D_BLOCK + 255) / 256, 256, 0, stream>>>(TW1, TW1p, D_MAIN, D_BLOCK);

  k_encode<<<NC / 16, 256, 0, stream>>>(cxn, linWp, linb, KWp, Kb,
                                        cand_k, c_sq, nullptr);
  k_encode<<<BQ / 16, 256, 0, stream>>>(x_num, linWp, linb, KWp, Kb,
                                        xkbuf, k_sq, xhbuf);
  k_knn<<<dim3(BQ / 16, PSEG), 256, 0, stream>>>(xkbuf, k_sq, cand_k, c_sq,
                                                 part_v, part_i);
  k_merge<<<BQ, 32, 0, stream>>>(part_v, part_i, top_v, top_i);
  k_final<<<BQ, 256, 0, stream>>>(xhbuf, xkbuf, top_v, top_i, candy, cand_k,
                                  yew, yeb, TW1p, Tb1, TW2,
                                  bgp, bbe, bW1, bb1, bW2, bb2,
                                  hg, hbe, hW, hb, (float*)d_out);
}